// Pllay_61873298866716
// MI455X (gfx1250) — compile-verified
//
#include <hip/hip_runtime.h>
#include <hip/hip_bf16.h>
#include <float.h>
#include <math.h>

typedef __attribute__((ext_vector_type(2))) float v2f;
typedef __attribute__((ext_vector_type(8))) float v8f;

#define HGRID 28
#define NV 784
#define E_CNT 1512
#define ESORT 2048
#define T_CNT 25
#define M0_F 0.05f
#define NOFF 3025
#define NOFF_PAD 4096

// ---- workspace byte offsets ----
#define WS_OFF_SORT 0        // 3025 u32 sorted offset keys  (reserve 16 KB)
#define WS_FEATS   16384     // 128*64 f32 padded features
#define WS_W1      49152     // 64*112 f32 padded w_topo
#define WS_B1      77824     // 112 f32 padded b_topo
#define WS_W2      78336     // 112*16 f32 padded w_fc
#define WS_B2      85504     // 16 f32 padded b_fc
#define WS_HID     85760     // 128*112 f32 hidden activations

// ------------------------------------------------------------------
// Kernel 1: build the universal sorted relative-offset table.
// key = (di^2+dj^2) << 12 | offsetIndex ; bitonic sort 4096 in LDS.
// ------------------------------------------------------------------
__global__ __launch_bounds__(256) void setup_offsets_kernel(unsigned int* offs) {
    __shared__ unsigned int key[NOFF_PAD];
    int tid = threadIdx.x;
    for (int i = tid; i < NOFF_PAD; i += 256) {
        if (i < NOFF) {
            int di = i / 55 - 27, dj = i % 55 - 27;
            unsigned int d2 = (unsigned int)(di * di + dj * dj);  // <= 1458
            key[i] = (d2 << 12) | (unsigned int)i;
        } else {
            key[i] = 0xFFFFFFFFu;
        }
    }
    __syncthreads();
    for (int k = 2; k <= NOFF_PAD; k <<= 1) {
        for (int j = k >> 1; j > 0; j >>= 1) {
            for (int i = tid; i < NOFF_PAD; i += 256) {
                int ixj = i ^ j;
                if (ixj > i) {
                    bool up = ((i & k) == 0);
                    unsigned int a = key[i], b = key[ixj];
                    if (up ? (a > b) : (a < b)) { key[i] = b; key[ixj] = a; }
                }
            }
            __syncthreads();
        }
    }
    for (int i = tid; i < NOFF; i += 256) offs[i] = key[i];
}

// ------------------------------------------------------------------
// Kernel 2: zero-pad MLP weights for full WMMA tiles.
// ------------------------------------------------------------------
__global__ __launch_bounds__(256) void prep_weights_kernel(
    const float* __restrict__ w_topo, const float* __restrict__ b_topo,
    const float* __restrict__ w_fc,   const float* __restrict__ b_fc,
    float* w1p, float* b1p, float* w2p, float* b2p) {
    int i = blockIdx.x * blockDim.x + threadIdx.x;
    if (i < 64 * 112) {
        int k = i / 112, n = i % 112;
        w1p[i] = (k < 50 && n < 100) ? w_topo[k * 100 + n] : 0.0f;
    }
    if (i < 112) b1p[i] = (i < 100) ? b_topo[i] : 0.0f;
    if (i < 112 * 16) {
        int k = i / 16, n = i % 16;
        w2p[i] = (k < 100 && n < 10) ? w_fc[k * 10 + n] : 0.0f;
    }
    if (i < 16) b2p[i] = (i < 10) ? b_fc[i] : 0.0f;
}

// ------------------------------------------------------------------
// Kernel 3: fused per-batch topology pipeline, fully LDS-resident.
// DTM filtration -> edge sort (bitonic) -> Kruskal PH0 -> landscape.
// One 256-thread block (8 waves) per batch.
// ------------------------------------------------------------------
__global__ __launch_bounds__(256) void topo_kernel(
    const float* __restrict__ x, const unsigned int* __restrict__ offs,
    float* __restrict__ feats) {
    __shared__ unsigned int s_off[NOFF];
    __shared__ float s_m[NV];
    __shared__ float s_f[NV];
    __shared__ float s_key[ESORT];
    __shared__ int   s_idx[ESORT];
    __shared__ int   s_par[NV];
    __shared__ float s_bs[E_CNT];
    __shared__ float s_ds[E_CNT];
    __shared__ float s_red[256];

    int tid = threadIdx.x;
    int b = blockIdx.x;
    const float* w = x + b * NV;

    for (int i = tid; i < NOFF; i += 256) s_off[i] = offs[i];
    float lsum = 0.0f;
    for (int v = tid; v < NV; v += 256) { float t = w[v]; s_m[v] = t; lsum += t; }
    s_red[tid] = lsum;
    __syncthreads();
    for (int s = 128; s > 0; s >>= 1) {
        if (tid < s) s_red[tid] += s_red[tid + s];
        __syncthreads();
    }
    float inv = 1.0f / fmaxf(s_red[0], 1e-12f);
    __syncthreads();
    for (int v = tid; v < NV; v += 256) s_m[v] *= inv;
    __syncthreads();

    // --- DTM walk: sorted offsets, early exit once M0 mass gathered ---
    for (int v = tid; v < NV; v += 256) {
        int vi = v / HGRID, vj = v % HGRID;
        float cum = 0.0f, acc = 0.0f;
        for (int jj = 0; jj < NOFF; ++jj) {
            unsigned int key = s_off[jj];
            int oi = (int)(key & 0xFFFu);
            int di = oi / 55 - 27, dj = oi % 55 - 27;
            int ui = vi + di, uj = vj + dj;
            if ((unsigned)ui < (unsigned)HGRID && (unsigned)uj < (unsigned)HGRID) {
                float msj = s_m[ui * HGRID + uj];
                float take = fminf(msj, fmaxf(M0_F - cum, 0.0f));
                float d2 = (float)(int)(key >> 12) * (1.0f / 729.0f);
                acc += take * d2;
                cum += msj;
                if (cum >= M0_F) break;
            }
        }
        s_f[v] = sqrtf(acc * (1.0f / M0_F) + 1e-12f);
    }
    __syncthreads();

    // --- edge weights (pack endpoints in payload), pad to 2048 ---
    for (int e = tid; e < ESORT; e += 256) {
        if (e < E_CNT) {
            int a, bb;
            if (e < 756) { int i = e / 27, j = e % 27; a = i * HGRID + j; bb = a + 1; }
            else         { int e2 = e - 756; a = e2; bb = e2 + HGRID; }
            s_key[e] = fmaxf(s_f[a], s_f[bb]);
            s_idx[e] = (a << 16) | bb;
        } else { s_key[e] = FLT_MAX; s_idx[e] = -1; }
    }
    for (int v = tid; v < NV; v += 256) s_par[v] = v;
    __syncthreads();

    // --- bitonic sort edges ascending by weight ---
    for (int k = 2; k <= ESORT; k <<= 1) {
        for (int j = k >> 1; j > 0; j >>= 1) {
            for (int i = tid; i < ESORT; i += 256) {
                int ixj = i ^ j;
                if (ixj > i) {
                    bool up = ((i & k) == 0);
                    float ka = s_key[i], kb = s_key[ixj];
                    if (up ? (ka > kb) : (ka < kb)) {
                        s_key[i] = kb; s_key[ixj] = ka;
                        int ta = s_idx[i]; s_idx[i] = s_idx[ixj]; s_idx[ixj] = ta;
                    }
                }
            }
            __syncthreads();
        }
    }

    // --- Kruskal / elder rule (sequential; 1512 edges) ---
    if (tid == 0) {
        for (int k = 0; k < E_CNT; ++k) {
            int pk = s_idx[k];
            int a = pk >> 16, bb = pk & 0xFFFF;
            int ra = a; while (s_par[ra] != ra) { s_par[ra] = s_par[s_par[ra]]; ra = s_par[ra]; }
            int rb = bb; while (s_par[rb] != rb) { s_par[rb] = s_par[s_par[rb]]; rb = s_par[rb]; }
            if (ra != rb) {
                float fa = s_f[ra], fb = s_f[rb];
                int young = (fa >= fb) ? ra : rb;
                int old   = (fa >= fb) ? rb : ra;
                s_bs[k] = s_f[young];
                s_ds[k] = s_key[k];
                s_par[young] = old;
            } else { s_bs[k] = 1e30f; s_ds[k] = -1e30f; }  // tent == 0
        }
    }
    __syncthreads();

    // --- persistence landscape: top-2 tents at 25 t's ---
    if (tid < T_CNT) {
        float t = 0.5f * (float)tid / (float)(T_CNT - 1);
        float m1 = 0.0f, m2 = 0.0f;
        for (int e = 0; e < E_CNT; ++e) {
            float tent = fmaxf(fminf(t - s_bs[e], s_ds[e] - t), 0.0f);
            if (tent > m1) { m2 = m1; m1 = tent; }
            else if (tent > m2) { m2 = tent; }
        }
        feats[b * 64 + tid] = m1;            // k=0 slice
        feats[b * 64 + T_CNT + tid] = m2;    // k=1 slice
    } else if (tid >= 2 * T_CNT && tid < 64) {
        feats[b * 64 + tid] = 0.0f;          // K-pad
    }
}

// ------------------------------------------------------------------
// Kernel 4/5: f32 WMMA GEMMs (V_WMMA_F32_16X16X4_F32).
// One wave per 16x16 output tile; A/B fragments per ISA 7.12.2:
//   A 16x4: lanes 0-15 K={0,1}, lanes 16-31 K={2,3} (VGPR0/1)
//   B 4x16: mirrored along N. C/D: VGPR i -> M=i+8*(lane>=16), N=lane&15.
// ------------------------------------------------------------------
__global__ __launch_bounds__(32) void gemm1_kernel(
    const float* __restrict__ A, const float* __restrict__ B,
    const float* __restrict__ bias, float* __restrict__ Cout) {
    int mt = blockIdx.x / 7, nt = blockIdx.x % 7;
    int m0 = mt * 16, n0 = nt * 16;
    int lane = threadIdx.x;
    int half = lane >> 4, l = lane & 15;
    v8f acc = {};
    for (int k0 = 0; k0 < 64; k0 += 4) {
        int ka = k0 + 2 * half;
        v2f a, bf;
        a.x  = A[(m0 + l) * 64 + ka];
        a.y  = A[(m0 + l) * 64 + ka + 1];
        bf.x = B[ka * 112 + n0 + l];
        bf.y = B[(ka + 1) * 112 + n0 + l];
        acc = __builtin_amdgcn_wmma_f32_16x16x4_f32(false, a, false, bf,
                                                    (short)0, acc, false, false);
    }
    float bv = bias[n0 + l];
#pragma unroll
    for (int i = 0; i < 8; ++i) {
        int row = m0 + i + half * 8;
        Cout[row * 112 + n0 + l] = fmaxf(acc[i] + bv, 0.0f);   // fused bias+ReLU
    }
}

__global__ __launch_bounds__(32) void gemm2_kernel(
    const float* __restrict__ A, const float* __restrict__ B,
    const float* __restrict__ bias, float* __restrict__ Out) {
    int m0 = blockIdx.x * 16;
    int lane = threadIdx.x;
    int half = lane >> 4, l = lane & 15;
    v8f acc = {};
    for (int k0 = 0; k0 < 112; k0 += 4) {
        int ka = k0 + 2 * half;
        v2f a, bf;
        a.x  = A[(m0 + l) * 112 + ka];
        a.y  = A[(m0 + l) * 112 + ka + 1];
        bf.x = B[ka * 16 + l];
        bf.y = B[(ka + 1) * 16 + l];
        acc = __builtin_amdgcn_wmma_f32_16x16x4_f32(false, a, false, bf,
                                                    (short)0, acc, false, false);
    }
    float bv = bias[l];
#pragma unroll
    for (int i = 0; i < 8; ++i) {
        int row = m0 + i + half * 8;
        if (l < 10) Out[row * 10 + l] = acc[i] + bv;
    }
}

extern "C" void kernel_launch(void* const* d_in, const int* in_sizes, int n_in,
                              void* d_out, int out_size, void* d_ws, size_t ws_size,
                              hipStream_t stream) {
    (void)in_sizes; (void)n_in; (void)out_size; (void)ws_size;
    const float* x      = (const float*)d_in[0];
    const float* w_topo = (const float*)d_in[1];
    const float* b_topo = (const float*)d_in[2];
    const float* w_fc   = (const float*)d_in[3];
    const float* b_fc   = (const float*)d_in[4];

    char* ws = (char*)d_ws;
    unsigned int* offs = (unsigned int*)(ws + WS_OFF_SORT);
    float* feats = (float*)(ws + WS_FEATS);
    float* w1p   = (float*)(ws + WS_W1);
    float* b1p   = (float*)(ws + WS_B1);
    float* w2p   = (float*)(ws + WS_W2);
    float* b2p   = (float*)(ws + WS_B2);
    float* hid   = (float*)(ws + WS_HID);

    setup_offsets_kernel<<<1, 256, 0, stream>>>(offs);
    prep_weights_kernel<<<28, 256, 0, stream>>>(w_topo, b_topo, w_fc, b_fc,
                                                w1p, b1p, w2p, b2p);
    topo_kernel<<<128, 256, 0, stream>>>(x, offs, feats);
    gemm1_kernel<<<56, 32, 0, stream>>>(feats, w1p, b1p, hid);
    gemm2_kernel<<<8, 32, 0, stream>>>(hid, w2p, b2p, (float*)d_out);
}